// Aggregate_14783277433345
// MI455X (gfx1250) — compile-verified
//
#include <hip/hip_runtime.h>
#include <hip/hip_bf16.h>
#include <stdint.h>

// Segment-sum over columns: out[b,g] = sum_{c: seg[c]==g} x[b,c]
// B = 8192, C = 8192, G = 512. Pure streaming problem: ~272MB of traffic,
// memory floor ~11.7us @ 23.3 TB/s. LDS-atomic scatter + async seg-id staging.

#define BATCH     8192
#define NCLASSES  8192
#define NGROUPS   512
#define ROWS_PER_BLOCK 8      // one wave32 per row
#define THREADS   256         // 8 waves

typedef float __attribute__((ext_vector_type(4))) v4f;
typedef int   __attribute__((ext_vector_type(4))) v4i;

__global__ __launch_bounds__(THREADS)
void Aggregate_14783277433345_kernel(const float* __restrict__ x,
                                     const int*   __restrict__ seg,
                                     float*       __restrict__ out) {
    __shared__ int   seg_lds[NCLASSES];                 // 32 KB, staged once, reused 8x
    __shared__ float acc[ROWS_PER_BLOCK][NGROUPS];      // 16 KB accumulators

    const int tid  = threadIdx.x;
    const int wave = tid >> 5;          // wave32: 8 waves per block
    const int lane = tid & 31;

    // ---- zero the LDS accumulators (flat, branch-free: 16 stores/thread) ----
    {
        float* af = &acc[0][0];
        #pragma unroll
        for (int k = 0; k < (ROWS_PER_BLOCK * NGROUPS) / THREADS; ++k) {
            af[tid + k * THREADS] = 0.0f;
        }
    }

    // ---- async stage segment ids into LDS (CDNA5 async global->LDS path) ----
    // 8192 ints = 32KB; 256 lanes x 16B = 4KB per round, 8 rounds.
    {
        const int*  gsrc     = seg + tid * 4;
        unsigned    lds_base = (unsigned)(uintptr_t)(&seg_lds[tid * 4]);
        #pragma unroll
        for (int r = 0; r < 8; ++r) {
            asm volatile("global_load_async_to_lds_b128 %0, %1, off"
                         :: "v"(lds_base + r * 4096),
                            "v"(gsrc + r * 1024)
                         : "memory");
        }
        asm volatile("s_wait_asynccnt 0" ::: "memory");
    }
    __syncthreads();

    // ---- main streaming loop: one row per wave ----
    const int row = blockIdx.x * ROWS_PER_BLOCK + wave;
    const v4f* __restrict__ xrow =
        (const v4f*)(x + (size_t)row * NCLASSES);   // 16B-aligned row
    float* __restrict__ arow = &acc[wave][0];

    // Per iteration: each lane handles 2 float4's (wave covers 256 columns).
    // Non-temporal loads: x is read exactly once, keep it out of L2.
    // unroll 2 => 4 b128 loads in flight per wave.
    #pragma unroll 2
    for (int c0 = 0; c0 < NCLASSES; c0 += 256) {
        const int i0 = (c0 >> 2) + lane;            // float4 index
        v4f a = __builtin_nontemporal_load(xrow + i0);
        v4f b = __builtin_nontemporal_load(xrow + i0 + 32);
        v4i sa = *(const v4i*)(&seg_lds[c0 + lane * 4]);         // ds_load_b128
        v4i sb = *(const v4i*)(&seg_lds[c0 + 128 + lane * 4]);   // ds_load_b128

        __hip_atomic_fetch_add(arow + sa.x, a.x, __ATOMIC_RELAXED, __HIP_MEMORY_SCOPE_WORKGROUP);
        __hip_atomic_fetch_add(arow + sa.y, a.y, __ATOMIC_RELAXED, __HIP_MEMORY_SCOPE_WORKGROUP);
        __hip_atomic_fetch_add(arow + sa.z, a.z, __ATOMIC_RELAXED, __HIP_MEMORY_SCOPE_WORKGROUP);
        __hip_atomic_fetch_add(arow + sa.w, a.w, __ATOMIC_RELAXED, __HIP_MEMORY_SCOPE_WORKGROUP);
        __hip_atomic_fetch_add(arow + sb.x, b.x, __ATOMIC_RELAXED, __HIP_MEMORY_SCOPE_WORKGROUP);
        __hip_atomic_fetch_add(arow + sb.y, b.y, __ATOMIC_RELAXED, __HIP_MEMORY_SCOPE_WORKGROUP);
        __hip_atomic_fetch_add(arow + sb.z, b.z, __ATOMIC_RELAXED, __HIP_MEMORY_SCOPE_WORKGROUP);
        __hip_atomic_fetch_add(arow + sb.w, b.w, __ATOMIC_RELAXED, __HIP_MEMORY_SCOPE_WORKGROUP);
    }
    __syncthreads();

    // ---- coalesced writeback: 8 rows x 512 groups per block ----
    {
        const float* af = &acc[0][0];
        float* op = out + (size_t)blockIdx.x * ROWS_PER_BLOCK * NGROUPS;
        #pragma unroll
        for (int k = 0; k < (ROWS_PER_BLOCK * NGROUPS) / THREADS; ++k) {
            __builtin_nontemporal_store(af[tid + k * THREADS], op + tid + k * THREADS);
        }
    }
}

extern "C" void kernel_launch(void* const* d_in, const int* in_sizes, int n_in,
                              void* d_out, int out_size, void* d_ws, size_t ws_size,
                              hipStream_t stream) {
    const float* x   = (const float*)d_in[0];   // [8192, 8192] fp32
    const int*   seg = (const int*)d_in[1];     // [8192] int32 in [0, 512)
    float*       out = (float*)d_out;           // [8192, 512] fp32

    const int grid = BATCH / ROWS_PER_BLOCK;    // 1024 blocks
    Aggregate_14783277433345_kernel<<<grid, THREADS, 0, stream>>>(x, seg, out);
}